// ConstraintLoss_38740605010347
// MI455X (gfx1250) — compile-verified
//
#include <hip/hip_runtime.h>
#include <hip/hip_bf16.h>

// ---------------------------------------------------------------------------
// ConstraintLoss for MI455X (gfx1250, wave32).
// Bandwidth-bound COO segment-sum: ~252MB traffic -> ~11us floor @ 23.3TB/s.
// No dense GEMM structure -> WMMA not applicable; optimize the memory system:
//   * double-buffered async global->LDS staging (ASYNCcnt path) for the
//     three 80MB streams; s_wait_asynccnt(6) keeps one tile in flight
//   * non-returning global_atomic_add_f32 scatter (ax stays L2-resident)
//   * hardware v_rcp_f32 sigmoid (avoid IEEE div chain)
//   * wave32 shuffle + ds_add_f32 reductions
// ---------------------------------------------------------------------------

typedef int   v4i __attribute__((ext_vector_type(4)));
typedef float v4f __attribute__((ext_vector_type(4)));

// Exact parameter types for __builtin_amdgcn_global_load_async_to_lds_b128,
// per hipcc diagnostic: non-const pointer to 16B int vector, as(1) src / as(3) dst.
typedef int b128_t __attribute__((vector_size(16)));
typedef __attribute__((address_space(1))) b128_t* g_b128_p;
typedef __attribute__((address_space(3))) b128_t* l_b128_p;

#define BLOCK 256
#define TILE  2048   // nnz elements per LDS tile (3 arrays * 8KB = 24KB per buffer)

#if defined(__HIP_DEVICE_COMPILE__) && defined(__gfx1250__) && \
    __has_builtin(__builtin_amdgcn_global_load_async_to_lds_b128) && \
    __has_builtin(__builtin_amdgcn_s_wait_asynccnt)
#define USE_ASYNC_LDS 1
#else
#define USE_ASYNC_LDS 0
#endif

__device__ __forceinline__ float fast_rcp(float x) {
#if defined(__HIP_DEVICE_COMPILE__)
    return __builtin_amdgcn_rcpf(x);   // v_rcp_f32
#else
    return 1.0f / x;
#endif
}

// ---- Kernel A: values = sigmoid(pred); ax = 0; out = 0 --------------------
__global__ __launch_bounds__(BLOCK)
void cl_init_kernel(const float* __restrict__ pred,
                    float* __restrict__ values,
                    float* __restrict__ ax,
                    float* __restrict__ out,
                    int n_vars, int n_constrs) {
    int i = blockIdx.x * BLOCK + threadIdx.x;
    if (i < n_vars) {
        float x = pred[i];
        values[i] = fast_rcp(1.0f + __expf(-x));
    }
    if (i < n_constrs) {
        ax[i] = 0.0f;
    }
    if (i == 0) {
        out[0] = 0.0f;
    }
}

// ---- Kernel B: ax[cidx[i]] += coeff[i] * values[vidx[i]] ------------------
__global__ __launch_bounds__(BLOCK)
void cl_scatter_kernel(const int* __restrict__ cidx,
                       const int* __restrict__ vidx,
                       const float* __restrict__ coeff,
                       const float* __restrict__ values,
                       float* __restrict__ ax,
                       long long nnz) {
#if USE_ASYNC_LDS
    // Double-buffered async staging: each thread copies its own 8 elements
    // per array (2 x b128 per array), then consumes exactly what it staged ->
    // per-wave s_wait_asynccnt suffices, no __syncthreads, no cross-wave
    // hazards. While tile i is processed, tile i+1 streams into the other
    // buffer (async loads complete in order: wait <=6 == "tile i done").
    __shared__ __align__(16) int   s_ci[2][TILE];
    __shared__ __align__(16) int   s_vi[2][TILE];
    __shared__ __align__(16) float s_co[2][TILE];

    const int t  = threadIdx.x;
    const int o0 = t * 8;
    const int o1 = o0 + 4;
    const long long ntiles = nnz / TILE;
    const long long tstep  = gridDim.x;

    auto issue = [&](long long tile, int buf) {
        const long long base = tile * (long long)TILE;
        __builtin_amdgcn_global_load_async_to_lds_b128(
            (g_b128_p)(cidx + base + o0), (l_b128_p)(&s_ci[buf][o0]), 0, 0);
        __builtin_amdgcn_global_load_async_to_lds_b128(
            (g_b128_p)(cidx + base + o1), (l_b128_p)(&s_ci[buf][o1]), 0, 0);
        __builtin_amdgcn_global_load_async_to_lds_b128(
            (g_b128_p)(vidx + base + o0), (l_b128_p)(&s_vi[buf][o0]), 0, 0);
        __builtin_amdgcn_global_load_async_to_lds_b128(
            (g_b128_p)(vidx + base + o1), (l_b128_p)(&s_vi[buf][o1]), 0, 0);
        __builtin_amdgcn_global_load_async_to_lds_b128(
            (g_b128_p)(coeff + base + o0), (l_b128_p)(&s_co[buf][o0]), 0, 0);
        __builtin_amdgcn_global_load_async_to_lds_b128(
            (g_b128_p)(coeff + base + o1), (l_b128_p)(&s_co[buf][o1]), 0, 0);
    };

    long long tile = blockIdx.x;
    int buf = 0;
    if (tile < ntiles) {
        issue(tile, 0);
    }
    for (; tile < ntiles; tile += tstep) {
        const long long nxt = tile + tstep;
        if (nxt < ntiles) {
            issue(nxt, buf ^ 1);
            __builtin_amdgcn_s_wait_asynccnt(6);  // current tile's 6 loads done
        } else {
            __builtin_amdgcn_s_wait_asynccnt(0);
        }
#pragma unroll
        for (int k = 0; k < 8; ++k) {
            const int idx = o0 + k;
            const int c = s_ci[buf][idx];
            const float val = values[s_vi[buf][idx]] * s_co[buf][idx];
            atomicAdd(&ax[c], val);   // non-returning global_atomic_add_f32
        }
        buf ^= 1;
    }

    // Tail (nnz not a multiple of TILE): plain scalar path.
    const long long done   = ntiles * (long long)TILE;
    const long long stride = (long long)gridDim.x * BLOCK;
    for (long long i = done + (long long)blockIdx.x * BLOCK + threadIdx.x;
         i < nnz; i += stride) {
        atomicAdd(&ax[cidx[i]], values[vidx[i]] * coeff[i]);
    }
#else
    // Fallback: vectorized NT streaming loads, grid-stride over quads.
    const long long nq = nnz >> 2;
    const long long stride = (long long)gridDim.x * BLOCK;
    for (long long q = (long long)blockIdx.x * BLOCK + threadIdx.x;
         q < nq; q += stride) {
        const long long i = q << 2;
        v4i ci = __builtin_nontemporal_load((const v4i*)(cidx + i));
        v4i vi = __builtin_nontemporal_load((const v4i*)(vidx + i));
        v4f co = __builtin_nontemporal_load((const v4f*)(coeff + i));
#pragma unroll
        for (int k = 0; k < 4; ++k) {
            atomicAdd(&ax[ci[k]], values[vi[k]] * co[k]);
        }
    }
    for (long long i = (nq << 2) + (long long)blockIdx.x * BLOCK + threadIdx.x;
         i < nnz; i += stride) {
        atomicAdd(&ax[cidx[i]], values[vidx[i]] * coeff[i]);
    }
#endif
}

// ---- Kernel C: violation + mean reduction (wave32 shuffles + ds_add_f32) --
__global__ __launch_bounds__(BLOCK)
void cl_violation_kernel(const float* __restrict__ ax,
                         const float* __restrict__ rhs,
                         const int* __restrict__ sense,
                         float* __restrict__ out,
                         int n_constrs, float inv_n) {
    const int i = blockIdx.x * BLOCK + threadIdx.x;
    float v = 0.0f;
    if (i < n_constrs) {
        const float d = ax[i] - rhs[i];
        const int s = sense[i];
        v = (s == 1) ? fmaxf(d, 0.0f)
          : (s == 2) ? fmaxf(-d, 0.0f)
          : (s == 3) ? fabsf(d)
                     : 0.0f;
        v *= inv_n;
    }
    // wave32 butterfly reduction
#pragma unroll
    for (int off = 16; off > 0; off >>= 1) {
        v += __shfl_down(v, off, 32);
    }
    __shared__ float bsum;
    if (threadIdx.x == 0) bsum = 0.0f;
    __syncthreads();
    if ((threadIdx.x & 31) == 0) {
        atomicAdd(&bsum, v);          // LDS ds_add_f32
    }
    __syncthreads();
    if (threadIdx.x == 0) {
        atomicAdd(out, bsum);         // one global atomic per block
    }
}

// ---------------------------------------------------------------------------
extern "C" void kernel_launch(void* const* d_in, const int* in_sizes, int n_in,
                              void* d_out, int out_size, void* d_ws, size_t ws_size,
                              hipStream_t stream) {
    const float* pred  = (const float*)d_in[0];
    const int*   cidx  = (const int*)d_in[1];
    const int*   vidx  = (const int*)d_in[2];
    const float* coeff = (const float*)d_in[3];
    const float* rhs   = (const float*)d_in[4];
    const int*   sense = (const int*)d_in[5];

    const int       n_vars    = in_sizes[0];
    const long long nnz       = in_sizes[1];
    const int       n_constrs = in_sizes[4];

    float* values = (float*)d_ws;            // n_vars floats
    float* ax     = values + n_vars;         // n_constrs floats
    float* out    = (float*)d_out;

    const int n_init = n_vars > n_constrs ? n_vars : n_constrs;
    cl_init_kernel<<<(n_init + BLOCK - 1) / BLOCK, BLOCK, 0, stream>>>(
        pred, values, ax, out, n_vars, n_constrs);

    // Grid-stride in both device-code variants; 8192 blocks saturates the chip.
    cl_scatter_kernel<<<8192, BLOCK, 0, stream>>>(
        cidx, vidx, coeff, values, ax, nnz);

    cl_violation_kernel<<<(n_constrs + BLOCK - 1) / BLOCK, BLOCK, 0, stream>>>(
        ax, rhs, sense, out, n_constrs, 1.0f / (float)n_constrs);
}